// EqLoss_49048526520785
// MI455X (gfx1250) — compile-verified
//
#include <hip/hip_runtime.h>
#include <hip/hip_bf16.h>
#include <stdint.h>

// ---------------------------------------------------------------------------
// EqLoss for MI455X (gfx1250):
//   * 512 MB logit stream -> HBM-bound (~22us floor at 23.3 TB/s).
//   * Data path: per-wave double-buffered GLOBAL_LOAD_ASYNC_TO_LDS_B128
//     (ASYNCcnt), 16x128 fp32 tile per wave, rows padded to 132 dwords so all
//     LDS reads are 64-bank conflict-free.
//   * Row sum-of-exp via chained V_WMMA_F32_16X16X4_F32 with B = ones
//     (exact fp32 accumulation, 32 WMMAs per 16x128 tile).
//   * Class segment-reduce: ds_add_f32 block accumulators -> one global
//     atomic per class per block; total loss accumulated in fp64.
// ---------------------------------------------------------------------------

typedef __attribute__((ext_vector_type(2))) float v2f;
typedef __attribute__((ext_vector_type(8))) float v8f;

#define ROW_BYTES   528                 // 128 floats + 16B pad (132 dwords)
#define TILE_BYTES  (16 * ROW_BYTES)    // 8448 B: one 16x128 tile
#define NBUF        16                  // 8 waves * 2 buffers
#define ACC_OFF     (NBUF * TILE_BYTES) // 135168
#define DYN_BYTES   (ACC_OFF + 1040)    // + blkSums[128] blkCnts[128] blkLoss
#define ALPHA_C     0.3f
#define EPS_C       1e-8f

#ifdef __has_builtin
#if __has_builtin(__builtin_amdgcn_wmma_f32_16x16x4_f32)
#define USE_WMMA 1
#endif
#endif

__device__ __forceinline__ void async_tile_load(const float* logits, int tile,
                                                unsigned ldsBase /* incl lane*16 */) {
  unsigned long long g0 = (unsigned long long)(uintptr_t)logits +
                          (unsigned long long)tile * (16ull * 128ull * 4ull) +
                          (unsigned long long)(ldsBase & 0u); // keep type simple
  g0 += 0; // base computed below per-lane
#pragma unroll
  for (int j = 0; j < 16; ++j) {
    unsigned long long ga = (unsigned long long)(uintptr_t)logits +
                            (unsigned long long)tile * (16ull * 128ull * 4ull) +
                            (unsigned long long)(unsigned)(j * 512) +
                            (unsigned long long)((ldsBase & 511u)); // lane*16 folded in ldsBase low bits? no
    (void)ga;
  }
}

// (helper above unused; real loader below keeps lane offset explicit)
__device__ __forceinline__ void tile_load_async(const float* logits, int tile,
                                                unsigned ldsBufBase, int lane) {
  unsigned long long gbase = (unsigned long long)(uintptr_t)logits +
                             (unsigned long long)tile * (16ull * 128ull * 4ull) +
                             (unsigned long long)(unsigned)(lane * 16);
  unsigned lbase = ldsBufBase + (unsigned)(lane * 16);
#pragma unroll
  for (int j = 0; j < 16; ++j) {
    unsigned long long ga = gbase + (unsigned long long)(unsigned)(j * 512);
    unsigned la = lbase + (unsigned)(j * ROW_BYTES);
    asm volatile("global_load_async_to_lds_b128 %0, %1, off"
                 :: "v"(la), "v"(ga)
                 : "memory");
  }
}

__global__ __launch_bounds__(256) void eq_main(const float* __restrict__ logits,
                                               const int* __restrict__ targets,
                                               float* __restrict__ gAcc,
                                               int numTiles) {
  extern __shared__ __align__(16) char smem[];
  const int tid  = threadIdx.x;
  const int lane = tid & 31;
  const int wave = tid >> 5;
  const int j    = lane & 15;   // row within 16-row tile
  const int h    = lane >> 4;   // half (K-split for WMMA A layout)

  float* blkSums = (float*)(smem + ACC_OFF);
  float* blkCnts = (float*)(smem + ACC_OFF + 512);
  float* blkLoss = (float*)(smem + ACC_OFF + 1024);
  for (int i = tid; i < 257; i += 256) ((float*)(smem + ACC_OFF))[i] = 0.f;
  __syncthreads();

  const int gwave  = blockIdx.x * 8 + wave;
  const int stride = gridDim.x * 8;
  const unsigned gss = __builtin_amdgcn_groupstaticsize();
  const unsigned bufBase0 = gss + (unsigned)((wave * 2 + 0) * TILE_BYTES);
  const unsigned bufBase1 = gss + (unsigned)((wave * 2 + 1) * TILE_BYTES);

  int tile = gwave;
  int buf  = 0;
  if (tile < numTiles) tile_load_async(logits, tile, bufBase0, lane);

  while (tile < numTiles) {
    const int nextTile = tile + stride;
    if (nextTile < numTiles) {
      tile_load_async(logits, nextTile, (buf ? bufBase0 : bufBase1), lane);
      asm volatile("s_wait_asynccnt 16" ::: "memory"); // current buffer done
    } else {
      asm volatile("s_wait_asynccnt 0" ::: "memory");
    }

    const char* wb = smem + (wave * 2 + buf) * TILE_BYTES;

    // -------- row max (each lane scans half of its row; 64-bank clean) ----
    const char* rowp = wb + j * ROW_BYTES + h * 256;
    float m = -3.402823466e38f;
#pragma unroll
    for (int i = 0; i < 16; ++i) {
      float4 v = *(const float4*)(rowp + i * 16);
      m = fmaxf(m, fmaxf(fmaxf(v.x, v.y), fmaxf(v.z, v.w)));
    }
    m = fmaxf(m, __shfl_xor(m, 16, 32));  // full-row max in both halves

    // -------- sum of exp(x - m) over 128 cols ----------------------------
    const char* ep = wb + j * ROW_BYTES + h * 8; // cols 4kc+2h, 4kc+2h+1
    float srow;
#ifdef USE_WMMA
    {
      v8f acc = {};
      const v2f bones = {1.0f, 1.0f};
#pragma unroll 4
      for (int kc = 0; kc < 32; ++kc) {
        float2 x = *(const float2*)(ep + kc * 16);
        v2f a;
        a.x = __expf(x.x - m);
        a.y = __expf(x.y - m);
        // D = A(16x4 f32) * ones(4x16) + C : accumulates per-row sums
        acc = __builtin_amdgcn_wmma_f32_16x16x4_f32(
            false, a, false, bones, (short)0, acc, false, false);
      }
      // D layout: VGPR v holds M=v (lanes 0-15) / M=v+8 (lanes 16-31);
      // every column identical, so pick element (j&7), then fetch from the
      // half that owns row j.
      float s_local;
      switch (j & 7) {
        case 0: s_local = acc[0]; break;
        case 1: s_local = acc[1]; break;
        case 2: s_local = acc[2]; break;
        case 3: s_local = acc[3]; break;
        case 4: s_local = acc[4]; break;
        case 5: s_local = acc[5]; break;
        case 6: s_local = acc[6]; break;
        default: s_local = acc[7]; break;
      }
      srow = __shfl(s_local, j | ((j & 8) << 1), 32);
    }
#else
    {
      float partial = 0.f;
#pragma unroll 4
      for (int kc = 0; kc < 32; ++kc) {
        float2 x = *(const float2*)(ep + kc * 16);
        partial += __expf(x.x - m) + __expf(x.y - m);
      }
      srow = partial + __shfl_xor(partial, 16, 32);
    }
#endif

    // -------- per-row loss + class accumulation (lanes 0..15) ------------
    float lv = 0.f;
    if (h == 0) {
      const int row = tile * 16 + j;
      const int t = targets[row];
      const float xt = *(const float*)(wb + j * ROW_BYTES + t * 4);
      const float loss = m + __logf(srow) - xt;
      atomicAdd(&blkSums[t], loss);   // ds_add_f32
      atomicAdd(&blkCnts[t], 1.0f);
      lv = loss;
    }
    lv += __shfl_xor(lv, 16, 32);
    lv += __shfl_xor(lv, 8, 32);
    lv += __shfl_xor(lv, 4, 32);
    lv += __shfl_xor(lv, 2, 32);
    lv += __shfl_xor(lv, 1, 32);
    if (lane == 0) atomicAdd(blkLoss, lv);

    buf ^= 1;
    tile = nextTile;
  }

  __syncthreads();
  if (tid < 128) {
    atomicAdd(&gAcc[tid], blkSums[tid]);
    atomicAdd(&gAcc[128 + tid], blkCnts[tid]);
  }
  if (tid == 0)
    atomicAdd((double*)((char*)gAcc + 1024), (double)blkLoss[0]);
}

// Scalar tail for N % 16 rows (not used for N = 1e6, kept for generality).
__global__ void eq_tail(const float* __restrict__ logits,
                        const int* __restrict__ targets,
                        float* __restrict__ gAcc, int start, int N) {
  int row = start + blockIdx.x * blockDim.x + threadIdx.x;
  if (row >= N) return;
  const float* x = logits + (size_t)row * 128;
  float m = x[0];
  for (int c = 1; c < 128; ++c) m = fmaxf(m, x[c]);
  float s = 0.f;
  for (int c = 0; c < 128; ++c) s += __expf(x[c] - m);
  int t = targets[row];
  float loss = m + __logf(s) - x[t];
  atomicAdd(&gAcc[t], loss);
  atomicAdd(&gAcc[128 + t], 1.0f);
  atomicAdd((double*)((char*)gAcc + 1024), (double)loss);
}

__global__ void eq_zero(float* ws) {
  int i = threadIdx.x;
  if (i < 258) ws[i] = 0.f;  // sums[128] + cnts[128] + double loss (2 words)
}

__global__ void eq_finalize(const float* __restrict__ gAcc, float* __restrict__ out,
                            float invN) {
  __shared__ float red[128];
  const int c = threadIdx.x;
  const float s = gAcc[c];
  const float n = gAcc[128 + c];
  const bool present = n > 0.f;
  const float cm = present ? (s / n) : 0.f;

  // nPresent
  red[c] = present ? 1.f : 0.f;
  __syncthreads();
  for (int off = 64; off > 0; off >>= 1) {
    if (c < off) red[c] += red[c + off];
    __syncthreads();
  }
  const float nP = red[0];
  __syncthreads();

  // sum of class means
  red[c] = cm;
  __syncthreads();
  for (int off = 64; off > 0; off >>= 1) {
    if (c < off) red[c] += red[c + off];
    __syncthreads();
  }
  const float cmMean = red[0] / nP;
  __syncthreads();

  // variance numerator
  const float d = present ? (cm - cmMean) : 0.f;
  red[c] = d * d;
  __syncthreads();
  for (int off = 64; off > 0; off >>= 1) {
    if (c < off) red[c] += red[c + off];
    __syncthreads();
  }

  if (c == 0) {
    const float var = red[0] / nP;
    const double total = *(const double*)((const char*)gAcc + 1024);
    const float mean = (float)(total * (double)invN);
    out[0] = mean + ALPHA_C * var / (cmMean + EPS_C);
  }
}

extern "C" void kernel_launch(void* const* d_in, const int* in_sizes, int n_in,
                              void* d_out, int out_size, void* d_ws, size_t ws_size,
                              hipStream_t stream) {
  const float* logits = (const float*)d_in[0];
  const int* targets  = (const int*)d_in[1];
  const int N = in_sizes[1];            // number of rows (targets count)
  const int numTiles = N / 16;
  const int rem = N % 16;
  float* ws = (float*)d_ws;

  (void)hipFuncSetAttribute((const void*)eq_main,
                            hipFuncAttributeMaxDynamicSharedMemorySize,
                            DYN_BYTES);

  eq_zero<<<1, 288, 0, stream>>>(ws);
  if (numTiles > 0) {
    eq_main<<<1024, 256, DYN_BYTES, stream>>>(logits, targets, ws, numTiles);
  }
  if (rem > 0) {
    eq_tail<<<1, 32, 0, stream>>>(logits, targets, ws, numTiles * 16, N);
  }
  eq_finalize<<<1, 128, 0, stream>>>(ws, (float*)d_out, 1.0f / (float)N);
}